// ExportableModel_5377299054767
// MI455X (gfx1250) — compile-verified
//
#include <hip/hip_runtime.h>
#include <math.h>

// ---------------------------------------------------------------------------
// Problem constants (match reference)
// ---------------------------------------------------------------------------
#define N_   16384
#define K_   16
#define EL_  (N_ * K_)
#define L_   2
#define NL_  3
#define CIN_ 170
#define H_   128
#define NEGF (-3.4028235e+38f)

typedef __attribute__((ext_vector_type(2))) float v2f;
typedef __attribute__((ext_vector_type(8))) float v8f;

// ---------------------------------------------------------------------------
// WMMA GEMM:  C[M x N] = A[M x K] * B[N x K]^T   (row-major A, row-major B)
// One wave32 per 16x16 C tile, V_WMMA_F32_16X16X4_F32 along K.
// mode 0: store   mode 1: BN(scale/bias from g,b,m,v per column)+ReLU store
// mode 2: C += result (accumulate)
// ---------------------------------------------------------------------------
__global__ __launch_bounds__(128)
void wmma_gemm_nt_f32(const float* __restrict__ A, const float* __restrict__ B,
                      float* __restrict__ C, int M, int N, int K,
                      const float* __restrict__ bng, const float* __restrict__ bnb,
                      const float* __restrict__ bnm, const float* __restrict__ bnv,
                      int mode) {
    const int lane = threadIdx.x & 31;
    const int wave = threadIdx.x >> 5;
    const int m0   = blockIdx.x * 16;
    const int n0   = (blockIdx.y * 4 + wave) * 16;
    const int half = lane >> 4;     // 0: K pair {0,1}, 1: K pair {2,3}
    const int r16  = lane & 15;     // row (A) / col (B) within tile

    const float* Arow = A + (size_t)(m0 + r16) * K;
    const float* Brow = B + (size_t)(n0 + r16) * K;

    v8f acc = {};
    int k = 0;
    for (; k + 3 < K; k += 4) {
        v2f a, b;
        const int kk = k + 2 * half;
        a.x = Arow[kk + 0];
        a.y = Arow[kk + 1];
        b.x = Brow[kk + 0];
        b.y = Brow[kk + 1];
        acc = __builtin_amdgcn_wmma_f32_16x16x4_f32(false, a, false, b,
                                                    (short)0, acc, false, false);
    }
    if (k < K) {  // ragged tail (K=170 for the layer-0 projections)
        v2f a = {}, b = {};
        const int kk = k + 2 * half;
        if (kk < K)     { a.x = Arow[kk];     b.x = Brow[kk];     }
        if (kk + 1 < K) { a.y = Arow[kk + 1]; b.y = Brow[kk + 1]; }
        acc = __builtin_amdgcn_wmma_f32_16x16x4_f32(false, a, false, b,
                                                    (short)0, acc, false, false);
    }

    // Epilogue. C/D layout: VGPR r -> row m0 + r + 8*half, col n0 + r16.
    const int col = n0 + r16;
    float scale = 1.0f, bias = 0.0f;
    if (mode == 1) {
        scale = bng[col] * rsqrtf(bnv[col] + 1e-5f);
        bias  = bnb[col] - bnm[col] * scale;
    }
#pragma unroll
    for (int r = 0; r < 8; ++r) {
        const int row = m0 + r + 8 * half;
        const size_t idx = (size_t)row * N + col;
        float v = acc[r];
        if (mode == 1) {
            v = v * scale + bias;
            v = v > 0.0f ? v : 0.0f;
            C[idx] = v;
        } else if (mode == 2) {
            C[idx] += v;
        } else {
            C[idx] = v;
        }
    }
}

// ---------------------------------------------------------------------------
// Edge phase. msg is NEVER materialized (EL*H*4B = 134MB); it is recomputed
// from the L2-resident src rows (N*H*4B = 8.4MB) in each pass:
//   msg[e,h] = relu(src[ei0[e],h] + ea[e]*we[h]) + 1e-7
// One wave per node; lane owns 4 contiguous channels (32*4 = H = 128).
// Node index is wave-uniform -> nbr/ei/ea loads become scalar loads.
// ---------------------------------------------------------------------------
__device__ __forceinline__ float4 msg4(const float* __restrict__ src,
                                       int srow, float a, float4 we, int h0) {
    const float4 s = *(const float4*)(src + (size_t)srow * H_ + h0);
    float4 v;
    v.x = fmaxf(s.x + a * we.x, 0.0f) + 1e-7f;
    v.y = fmaxf(s.y + a * we.y, 0.0f) + 1e-7f;
    v.z = fmaxf(s.z + a * we.z, 0.0f) + 1e-7f;
    v.w = fmaxf(s.w + a * we.w, 0.0f) + 1e-7f;
    return v;
}

// m[n,h] = max_k msg[nbr[n,k], h]
__global__ __launch_bounds__(256)
void scatter_max_kernel(const float* __restrict__ src, const int* __restrict__ ei0,
                        const float* __restrict__ ea, const float* __restrict__ we,
                        const int* __restrict__ nbr, int noff, int nstr,
                        float* __restrict__ mout) {
    const int lane = threadIdx.x & 31;
    const int n    = blockIdx.x * 8 + (threadIdx.x >> 5);
    const int h0   = lane * 4;
    const float4 wev = *(const float4*)(we + h0);
    const int* nrow = nbr + (size_t)n * nstr + noff;
    float4 mx = make_float4(NEGF, NEGF, NEGF, NEGF);
#pragma unroll 4
    for (int k = 0; k < K_; ++k) {
        const int e = nrow[k];
        const float4 v = msg4(src, ei0[e], ea[e], wev, h0);
        mx.x = fmaxf(mx.x, v.x); mx.y = fmaxf(mx.y, v.y);
        mx.z = fmaxf(mx.z, v.z); mx.w = fmaxf(mx.w, v.w);
    }
    *(float4*)(mout + (size_t)n * H_ + h0) = mx;
}

// s[n,h] = 1e-16 + sum_k exp(msg[e,h] - m[ei1[e],h]),  e = nbr[n,k]
__global__ __launch_bounds__(256)
void scatter_sumexp_kernel(const float* __restrict__ src, const int* __restrict__ ei0,
                           const int* __restrict__ ei1, const float* __restrict__ ea,
                           const float* __restrict__ we, const int* __restrict__ nbr,
                           int noff, int nstr, const float* __restrict__ m,
                           float* __restrict__ sout) {
    const int lane = threadIdx.x & 31;
    const int n    = blockIdx.x * 8 + (threadIdx.x >> 5);
    const int h0   = lane * 4;
    const float4 wev = *(const float4*)(we + h0);
    const int* nrow = nbr + (size_t)n * nstr + noff;
    float4 ss = make_float4(1e-16f, 1e-16f, 1e-16f, 1e-16f);
#pragma unroll 4
    for (int k = 0; k < K_; ++k) {
        const int e = nrow[k];
        const float4 v  = msg4(src, ei0[e], ea[e], wev, h0);
        const float4 mv = *(const float4*)(m + (size_t)ei1[e] * H_ + h0);
        ss.x += __expf(v.x - mv.x); ss.y += __expf(v.y - mv.y);
        ss.z += __expf(v.z - mv.z); ss.w += __expf(v.w - mv.w);
    }
    *(float4*)(sout + (size_t)n * H_ + h0) = ss;
}

// out[n,h] = dst[n,h] + sum_k msg[e,h] * exp(msg[e,h]-m[d,h]) / s[d,h]
__global__ __launch_bounds__(256)
void scatter_agg_kernel(const float* __restrict__ src, const int* __restrict__ ei0,
                        const int* __restrict__ ei1, const float* __restrict__ ea,
                        const float* __restrict__ we, const int* __restrict__ nbr,
                        int noff, int nstr, const float* __restrict__ m,
                        const float* __restrict__ s, const float* __restrict__ dst,
                        float* __restrict__ out) {
    const int lane = threadIdx.x & 31;
    const int n    = blockIdx.x * 8 + (threadIdx.x >> 5);
    const int h0   = lane * 4;
    const float4 wev = *(const float4*)(we + h0);
    const int* nrow = nbr + (size_t)n * nstr + noff;
    float4 acc = make_float4(0.f, 0.f, 0.f, 0.f);
#pragma unroll 4
    for (int k = 0; k < K_; ++k) {
        const int e = nrow[k];
        const int d = ei1[e];
        const float4 v  = msg4(src, ei0[e], ea[e], wev, h0);
        const float4 mv = *(const float4*)(m + (size_t)d * H_ + h0);
        const float4 sv = *(const float4*)(s + (size_t)d * H_ + h0);
        acc.x += v.x * __expf(v.x - mv.x) / sv.x;
        acc.y += v.y * __expf(v.y - mv.y) / sv.y;
        acc.z += v.z * __expf(v.z - mv.z) / sv.z;
        acc.w += v.w * __expf(v.w - mv.w) / sv.w;
    }
    const float4 dv = *(const float4*)(dst + (size_t)n * H_ + h0);
    float4 o;
    o.x = acc.x + dv.x; o.y = acc.y + dv.y;
    o.z = acc.z + dv.z; o.w = acc.w + dv.w;
    *(float4*)(out + (size_t)n * H_ + h0) = o;
}

__global__ __launch_bounds__(256)
void leaky_kernel(const float* __restrict__ y, float* __restrict__ h, int count) {
    const int i = blockIdx.x * blockDim.x + threadIdx.x;
    if (i < count) {
        const float v = y[i];
        h[i] = v > 0.0f ? v : 0.01f * v;
    }
}

// ---------------------------------------------------------------------------
// Host-side orchestration (graph-capture safe: launches only, all on stream)
// ---------------------------------------------------------------------------
extern "C" void kernel_launch(void* const* d_in, const int* in_sizes, int n_in,
                              void* d_out, int out_size, void* d_ws, size_t ws_size,
                              hipStream_t stream) {
    const float* x      = (const float*)d_in[0];   // (N, CIN)
    const int*   ei     = (const int*)  d_in[1];   // (2, L*EL)
    const float* ea     = (const float*)d_in[2];   // (L*EL, 1)
    const int*   nbr    = (const int*)  d_in[3];   // (N, L*K)
    // d_in[4] = size_idx (unused)
    const float* w_src0 = (const float*)d_in[5];   // (L, H, CIN)
    const float* w_dst0 = (const float*)d_in[6];   // (L, H, CIN)
    const float* w_edge = (const float*)d_in[7];   // (NL, L, H, 1)
    const float* w1     = (const float*)d_in[8];   // (NL, L, 2H, H)
    const float* bng    = (const float*)d_in[9];   // (NL, L, 2H)
    const float* bnb    = (const float*)d_in[10];
    const float* bnm    = (const float*)d_in[11];
    const float* bnv    = (const float*)d_in[12];
    const float* w2     = (const float*)d_in[13];  // (NL, L, H, 2H)
    float* out = (float*)d_out;                    // (N, H)

    // Workspace carve-out (floats).  Total ~76 MB; all hot tensors L2-resident.
    const size_t NH = (size_t)N_ * H_;
    float* ws_f   = (float*)d_ws;
    float* srcbuf = ws_f;              // N*H
    float* dstbuf = srcbuf + NH;       // N*H
    float* mbuf   = dstbuf + NH;       // N*H
    float* sbuf   = mbuf   + NH;       // N*H
    float* aggbuf = sbuf   + NH;       // N*H
    float* ybuf   = aggbuf + NH;       // N*H
    float* hbuf   = ybuf   + NH;       // N*H
    float* h1buf  = hbuf   + NH;       // N*2H

    const dim3 blk128(128), blk256(256);
    const dim3 grid_nodes(N_ / 8);                    // 8 waves/block, 1 node/wave

    for (int i = 0; i < NL_; ++i) {
        for (int l = 0; l < L_; ++l) {
            const float *src, *dst;
            if (i == 0) {
                // Layer-0 input projections: (N x 170) * (128 x 170)^T
                wmma_gemm_nt_f32<<<dim3(N_ / 16, H_ / 64), blk128, 0, stream>>>(
                    x, w_src0 + (size_t)l * H_ * CIN_, srcbuf, N_, H_, CIN_,
                    nullptr, nullptr, nullptr, nullptr, 0);
                wmma_gemm_nt_f32<<<dim3(N_ / 16, H_ / 64), blk128, 0, stream>>>(
                    x, w_dst0 + (size_t)l * H_ * CIN_, dstbuf, N_, H_, CIN_,
                    nullptr, nullptr, nullptr, nullptr, 0);
                src = srcbuf; dst = dstbuf;
            } else {
                src = hbuf; dst = hbuf;
            }

            const int*   ei0 = ei + (size_t)l * EL_;
            const int*   ei1 = ei + (size_t)L_ * EL_ + (size_t)l * EL_;
            const float* eal = ea + (size_t)l * EL_;
            const float* wel = w_edge + (size_t)(i * L_ + l) * H_;
            const int noff = l * K_, nstr = L_ * K_;

            scatter_max_kernel<<<grid_nodes, blk256, 0, stream>>>(
                src, ei0, eal, wel, nbr, noff, nstr, mbuf);
            scatter_sumexp_kernel<<<grid_nodes, blk256, 0, stream>>>(
                src, ei0, ei1, eal, wel, nbr, noff, nstr, mbuf, sbuf);
            scatter_agg_kernel<<<grid_nodes, blk256, 0, stream>>>(
                src, ei0, ei1, eal, wel, nbr, noff, nstr, mbuf, sbuf, dst, aggbuf);

            // h1 = relu(BN(agg @ w1^T)) : (N x 128) * (256 x 128)^T
            const size_t bno = (size_t)(i * L_ + l) * 2 * H_;
            wmma_gemm_nt_f32<<<dim3(N_ / 16, (2 * H_) / 64), blk128, 0, stream>>>(
                aggbuf, w1 + (size_t)(i * L_ + l) * 2 * H_ * H_, h1buf,
                N_, 2 * H_, H_, bng + bno, bnb + bno, bnm + bno, bnv + bno, 1);

            // y (+)= h1 @ w2^T : (N x 256) * (128 x 256)^T
            float* yout = (i == NL_ - 1) ? out : ybuf;
            wmma_gemm_nt_f32<<<dim3(N_ / 16, H_ / 64), blk128, 0, stream>>>(
                h1buf, w2 + (size_t)(i * L_ + l) * H_ * 2 * H_, yout,
                N_, H_, 2 * H_, nullptr, nullptr, nullptr, nullptr,
                (l == 0) ? 0 : 2);
        }
        if (i < NL_ - 1) {
            leaky_kernel<<<(int)((NH + 255) / 256), blk256, 0, stream>>>(
                ybuf, hbuf, (int)NH);
        }
    }
}